// Gate_78099685310873
// MI455X (gfx1250) — compile-verified
//
#include <hip/hip_runtime.h>
#include <math.h>

#define T_TOK 8192
#define DDIM  7168
#define NEXP  256
#define TOPK  8
#define NGRP  8
#define LGRP  4
#define GSZ   (NEXP / NGRP)     // 32 experts per group
#define ROUTE_SCALE 2.5f
#define NEG_INF (-3.4e38f)

#define MTILE 32                // tokens per block
#define CHUNK 128               // K-chunk staged in LDS per TDM op
#define NCHUNK (DDIM / CHUNK)   // 56

typedef float v2f __attribute__((ext_vector_type(2)));
typedef float v8f __attribute__((ext_vector_type(8)));
typedef unsigned int u32x4 __attribute__((ext_vector_type(4)));
typedef int i32x4 __attribute__((ext_vector_type(4)));
typedef int i32x8 __attribute__((ext_vector_type(8)));

__global__ void moe_zero_counts_kernel(int* __restrict__ counts) {
    counts[threadIdx.x] = 0;
}

// Build TDM descriptor and issue the tensor load (6-arg toolchain form).
// D# per CDNA5 ISA:
//   g0: count=1 | lds_addr[63:32] | global_addr[120:64] | type=2 [127:126]
//   g1: data_size=2(4B) [17:16] | tensor_dim0=D [79:48] | tensor_dim1=T [111:80]
//       | tile_dim0=CHUNK [127:112] | tile_dim1=MTILE [143:128]
//       | tensor_dim0_stride=D [207:160]
__device__ __forceinline__ void tdm_load_chunk(unsigned long long gaddr,
                                               unsigned int lds_off) {
    u32x4 g0;
    g0.x = 0x1u;                                   // count=1, user mode
    g0.y = lds_off;                                // lds_addr (bytes)
    g0.z = (unsigned int)(gaddr & 0xFFFFFFFFull);  // global_addr[31:0]
    g0.w = (unsigned int)((gaddr >> 32) & 0x01FFFFFFull) | (2u << 30); // [56:32] | type=2

    i32x8 g1;
    g1[0] = (2 << 16);                             // data_size = 2 -> 4 bytes
    g1[1] = (int)((DDIM & 0xFFFF) << 16);          // tensor_dim0 lo16 @ bits 63:48
    g1[2] = (int)(((DDIM >> 16) & 0xFFFF) | ((T_TOK & 0xFFFF) << 16)); // dim0 hi | dim1 lo
    g1[3] = (int)(((T_TOK >> 16) & 0xFFFF) | (CHUNK << 16));           // dim1 hi | tile_dim0
    g1[4] = MTILE;                                 // tile_dim1 = 32 (tile_dim2 = 0)
    g1[5] = DDIM;                                  // tensor_dim0_stride lo32
    g1[6] = 0;
    g1[7] = 0;

    i32x4 gz4 = {0, 0, 0, 0};
    i32x8 gz8 = {0, 0, 0, 0, 0, 0, 0, 0};
    __builtin_amdgcn_tensor_load_to_lds(g0, g1, gz4, gz4, gz8, 0);
}

__global__ __launch_bounds__(256)
void moe_gate_kernel(const float* __restrict__ x,
                     const float* __restrict__ W,
                     const float* __restrict__ bias,
                     float* __restrict__ outW,
                     long long* __restrict__ outI,
                     int* __restrict__ counts)
{
    __shared__ float sA[2][MTILE * CHUNK];     // double-buffered x slab (2 x 16 KB)
    __shared__ float sScore[MTILE][NEXP];      // sigmoid(logit) (original scores)
    __shared__ float sMask[MTILE][NEXP];       // biased / masked scores for selection
    __shared__ float sBias[NEXP];

    const int tid  = threadIdx.x;
    const int wave = tid >> 5;                 // 0..7 (wave32)
    const int lane = tid & 31;
    const int l16  = lane & 15;
    const int hlf  = lane >> 4;                // 0: K=0,1  1: K=2,3 (fp32 A/B layout)
    const int rowBase = blockIdx.x * MTILE;

    sBias[tid] = bias[tid];

    const unsigned long long xTileBase =
        (unsigned long long)(uintptr_t)x + (unsigned long long)rowBase * DDIM * 4ull;
    const unsigned int ldsA0 = (unsigned int)(uintptr_t)&sA[0][0];
    const unsigned int ldsA1 = (unsigned int)(uintptr_t)&sA[1][0];

    // ---- GEMM: 32x256 logits tile, A staged in LDS by the Tensor Data Mover ----
    const int col0 = wave * 32 + l16;          // first 16x16 N-tile for this wave
    const int col1 = col0 + 16;                // second N-tile
    const float* bPtr0 = W + (size_t)col0 * DDIM + 2 * hlf;
    const float* bPtr1 = W + (size_t)col1 * DDIM + 2 * hlf;

    v8f acc00 = {}, acc01 = {}, acc10 = {}, acc11 = {};

    // prologue: kick off chunk 0
    if (wave == 0) {
        tdm_load_chunk(xTileBase, ldsA0);
    }

    for (int c = 0; c < NCHUNK; ++c) {
        const int buf = c & 1;
        if (wave == 0) {
            if (c + 1 < NCHUNK) {
                // issue next chunk into the other buffer, then wait for chunk c
                tdm_load_chunk(xTileBase + (unsigned long long)(c + 1) * CHUNK * 4ull,
                               (c + 1) & 1 ? ldsA1 : ldsA0);
                __builtin_amdgcn_s_wait_tensorcnt(1);   // in-order: chunk c done
            } else {
                __builtin_amdgcn_s_wait_tensorcnt(0);
            }
        }
        __syncthreads();   // A chunk visible to all waves

        const float* aRow0 = &sA[buf][(size_t)l16 * CHUNK + 2 * hlf];        // rows 0..15
        const float* aRow1 = &sA[buf][(size_t)(l16 + 16) * CHUNK + 2 * hlf]; // rows 16..31
        const float* bC0   = bPtr0 + (size_t)c * CHUNK;
        const float* bC1   = bPtr1 + (size_t)c * CHUNK;

        #pragma unroll 4
        for (int kk = 0; kk < CHUNK; kk += 4) {
            v2f a0 = *(const v2f*)(aRow0 + kk);
            v2f a1 = *(const v2f*)(aRow1 + kk);
            v2f b0 = *(const v2f*)(bC0 + kk);
            v2f b1 = *(const v2f*)(bC1 + kk);
            acc00 = __builtin_amdgcn_wmma_f32_16x16x4_f32(false, a0, false, b0,
                                                          (short)0, acc00, false, false);
            acc01 = __builtin_amdgcn_wmma_f32_16x16x4_f32(false, a0, false, b1,
                                                          (short)0, acc01, false, false);
            acc10 = __builtin_amdgcn_wmma_f32_16x16x4_f32(false, a1, false, b0,
                                                          (short)0, acc10, false, false);
            acc11 = __builtin_amdgcn_wmma_f32_16x16x4_f32(false, a1, false, b1,
                                                          (short)0, acc11, false, false);
        }
        __syncthreads();   // all waves done reading buf before it is re-targeted
    }

    // C/D layout: VGPR r holds (M = r + 8*hlf, N = l16)
    #pragma unroll
    for (int r = 0; r < 8; ++r) {
        const int m = r + 8 * hlf;
        sScore[m][wave * 32 + l16]           = acc00[r];
        sScore[m][wave * 32 + 16 + l16]      = acc01[r];
        sScore[m + 16][wave * 32 + l16]      = acc10[r];
        sScore[m + 16][wave * 32 + 16 + l16] = acc11[r];
    }
    __syncthreads();

    // ---- sigmoid + bias (parallel over the 32x256 tile) ----
    for (int i = tid; i < MTILE * NEXP; i += 256) {
        const int m = i >> 8;
        const int e = i & (NEXP - 1);
        const float s = 1.0f / (1.0f + __expf(-sScore[m][e]));
        sScore[m][e] = s;                // original score (for weights)
        sMask[m][e]  = s + sBias[e];     // biased score (for routing)
    }
    __syncthreads();

    // ---- per-token routing: one lane per token (one full wave) ----
    if (tid < MTILE) {
        const int m = tid;
        const long long token = (long long)(rowBase + m);

        // group score = sum of top-2 biased scores in each group of 32
        float gs[NGRP];
        #pragma unroll
        for (int g = 0; g < NGRP; ++g) {
            float m1 = NEG_INF, m2 = NEG_INF;
            for (int e = g * GSZ; e < (g + 1) * GSZ; ++e) {
                const float v = sMask[m][e];
                if (v > m1) { m2 = m1; m1 = v; }
                else if (v > m2) { m2 = v; }
            }
            gs[g] = m1 + m2;
        }

        // keep top-LGRP groups (strict > keeps first occurrence, like top_k)
        unsigned keep = 0;
        #pragma unroll
        for (int s = 0; s < LGRP; ++s) {
            int bg = 0; float bv = NEG_INF;
            #pragma unroll
            for (int g = 0; g < NGRP; ++g) {
                if (!((keep >> g) & 1u) && gs[g] > bv) { bv = gs[g]; bg = g; }
            }
            keep |= (1u << bg);
        }

        // mask out non-kept groups
        for (int g = 0; g < NGRP; ++g) {
            if (!((keep >> g) & 1u)) {
                for (int e = g * GSZ; e < (g + 1) * GSZ; ++e)
                    sMask[m][e] = NEG_INF;
            }
        }

        // top-K experts by repeated argmax (descending order, first-tie wins)
        int   idx[TOPK];
        float wv[TOPK];
        float wsum = 0.0f;
        #pragma unroll
        for (int k = 0; k < TOPK; ++k) {
            int be = 0; float bv = NEG_INF;
            for (int e = 0; e < NEXP; ++e) {
                const float v = sMask[m][e];
                if (v > bv) { bv = v; be = e; }
            }
            sMask[m][be] = NEG_INF;
            idx[k] = be;
            wv[k]  = sScore[m][be];      // un-biased sigmoid score
            wsum  += wv[k];
        }

        const float inv = ROUTE_SCALE / wsum;
        #pragma unroll
        for (int k = 0; k < TOPK; ++k) {
            outW[token * TOPK + k] = wv[k] * inv;
            outI[token * TOPK + k] = (long long)idx[k];
            atomicAdd(&counts[idx[k]], 1);
        }
    }
}

extern "C" void kernel_launch(void* const* d_in, const int* in_sizes, int n_in,
                              void* d_out, int out_size, void* d_ws, size_t ws_size,
                              hipStream_t stream) {
    (void)in_sizes; (void)n_in; (void)out_size; (void)d_ws; (void)ws_size;

    const float* x    = (const float*)d_in[0];   // [T, D]
    const float* W    = (const float*)d_in[1];   // [E, D]
    const float* bias = (const float*)d_in[2];   // [E]

    // Outputs packed byte-wise in return order:
    //   weights [T,K] f32 | indices [T,K] i64 | counts [E] i32
    char* base = (char*)d_out;
    float*     outW   = (float*)base;
    long long* outI   = (long long*)(base + (size_t)T_TOK * TOPK * sizeof(float));
    int*       counts = (int*)(base + (size_t)T_TOK * TOPK * (sizeof(float) + sizeof(long long)));

    moe_zero_counts_kernel<<<1, NEXP, 0, stream>>>(counts);
    moe_gate_kernel<<<T_TOK / MTILE, 256, 0, stream>>>(x, W, bias, outW, outI, counts);
}